// S4Layer_81140522156898
// MI455X (gfx1250) — compile-verified
//
#include <hip/hip_runtime.h>

// S4/Mamba-style layer for MI455X (gfx1250, wave32).
//   Phase 1: dt = clamp(exp(u @ W_dt^T + b_dt), 1e-3, 2e-2)   -- f32 WMMA GEMM
//   Phase 2: sequential selective scan over L, recomputing A_bar/Bu on the fly.

typedef __attribute__((ext_vector_type(2))) float v2f;
typedef __attribute__((ext_vector_type(8))) float v8f;

constexpr int BZ   = 2;      // batch
constexpr int LSEQ = 2048;   // sequence length
constexpr int H    = 1024;   // d_model
constexpr int NS   = 16;     // d_state
constexpr int MTOT = BZ * LSEQ;          // 4096 GEMM rows
constexpr float DTMIN = 0.001f;
constexpr float DTMAX = 0.02f;

// ---------------------------------------------------------------------------
// Kernel 1: dt-projection GEMM via V_WMMA_F32_16X16X4_F32.
// One wave computes a 16(M) x 32(N) tile as two 16x16 accumulators sharing A.
// A fragment (16x4 f32): lane 0-15 -> M=lane, K = k+{0,1}; lane 16-31 -> K = k+{2,3}
// B fragment (4x16 f32): lane 0-15 -> N=lane, K = k+{0,1}; lane 16-31 -> K = k+{2,3}
//   B[k][n] = W_dt[n][k]  (W_dt is (g,h) row-major, i.e. already K-contiguous)
// C/D 16x16 f32: VGPR r, lane -> (M = r + 8*(lane>>4), N = lane&15)
// ---------------------------------------------------------------------------
__global__ __launch_bounds__(256) void s4_dtproj_wmma(
    const float* __restrict__ u,      // (MTOT, H)
    const float* __restrict__ W,      // (H, H)  row g contiguous in K
    const float* __restrict__ bvec,   // (H,)
    float* __restrict__ dt)           // (MTOT, H) workspace output
{
    const int K = H;
    const int lane = threadIdx.x & 31;
    const int gwid = blockIdx.x * (blockDim.x >> 5) + (threadIdx.x >> 5);

    const int tilesN = H / 32;               // 32 output cols per wave
    const int tm = gwid / tilesN;
    const int tn = gwid % tilesN;
    if (tm >= MTOT / 16) return;

    const int row0 = tm * 16;
    const int col0 = tn * 32;
    const int half = lane >> 4;              // 0: K-pair {0,1}, 1: K-pair {2,3}
    const int lm   = lane & 15;

    const float* __restrict__ aptr  = u + (size_t)(row0 + lm) * K + 2 * half;
    const float* __restrict__ bptr0 = W + (size_t)(col0 + lm) * K + 2 * half;
    const float* __restrict__ bptr1 = W + (size_t)(col0 + 16 + lm) * K + 2 * half;

    v8f acc0 = {};
    v8f acc1 = {};

    for (int k0 = 0; k0 < K; k0 += 16) {
        if (k0 + 16 < K) {                    // gfx1250 global_prefetch_b8
            __builtin_prefetch(aptr  + k0 + 16, 0, 3);
            __builtin_prefetch(bptr0 + k0 + 16, 0, 3);
            __builtin_prefetch(bptr1 + k0 + 16, 0, 3);
        }
#pragma unroll
        for (int kk = 0; kk < 16; kk += 4) {
            const int k = k0 + kk;
            v2f a  = *(const v2f*)(aptr  + k);
            v2f b0 = *(const v2f*)(bptr0 + k);
            v2f b1 = *(const v2f*)(bptr1 + k);
            // 8 args: (neg_a, A, neg_b, B, c_mod, C, reuse_a, reuse_b)
            acc0 = __builtin_amdgcn_wmma_f32_16x16x4_f32(
                false, a, false, b0, (short)0, acc0, false, false);
            acc1 = __builtin_amdgcn_wmma_f32_16x16x4_f32(
                false, a, false, b1, (short)0, acc1, false, false);
        }
    }

    // Fused epilogue: + bias, exp, clamp -> dt
    const int g0 = col0 + lm;
    const int g1 = col0 + 16 + lm;
    const float bb0 = bvec[g0];
    const float bb1 = bvec[g1];
#pragma unroll
    for (int r = 0; r < 8; ++r) {
        const int m = row0 + r + 8 * half;
        float d0 = __expf(acc0[r] + bb0);
        float d1 = __expf(acc1[r] + bb1);
        d0 = fminf(fmaxf(d0, DTMIN), DTMAX);
        d1 = fminf(fmaxf(d1, DTMIN), DTMAX);
        dt[(size_t)m * H + g0] = d0;
        dt[(size_t)m * H + g1] = d1;
    }
}

// ---------------------------------------------------------------------------
// Kernel 2: selective scan. One lane per (b, h, n); 16 lanes share one (b,h)
// recurrence and reduce y via intra-wave shuffles. A_bar and Bu are
// recomputed per step (expm1(z) = exp(z)-1), never materialized.
// ---------------------------------------------------------------------------
__global__ __launch_bounds__(256) void s4_scan(
    const float* __restrict__ u,      // (BZ, LSEQ, H)
    const float* __restrict__ A_log,  // (H, NS)
    const float* __restrict__ B_p,    // (H, NS)
    const float* __restrict__ C_p,    // (H, NS)
    const float* __restrict__ D_p,    // (H,)
    const float* __restrict__ dt,     // (BZ, LSEQ, H)
    float* __restrict__ y)            // (BZ, LSEQ, H)
{
    const int t = blockIdx.x * blockDim.x + threadIdx.x;
    const int n    = t & (NS - 1);
    const int pair = t >> 4;
    if (pair >= BZ * H) return;
    const int b = pair / H;
    const int h = pair - b * H;

    const float An   = -__expf(A_log[h * NS + n]);   // A = -exp(A_log)
    const float invA = __frcp_rn(An);
    const float Bn   = B_p[h * NS + n];
    const float Cn   = C_p[h * NS + n];
    const float Dh   = D_p[h];

    float x = 0.0f;
    const size_t base = (size_t)b * LSEQ * H + h;

    for (int l = 0; l < LSEQ; ++l) {
        const size_t idx = base + (size_t)l * H;
        const float d  = dt[idx];
        const float uu = u[idx];

        const float abar = __expf(d * An);                   // exp(dt*A)
        const float bu   = (abar - 1.0f) * invA * Bn * uu;   // expm1/A * B * u
        x = __builtin_fmaf(x, abar, bu);                     // x = x*A_bar + Bu

        float p = x * Cn;                                    // partial of C.x
        p += __shfl_xor(p, 8, 16);
        p += __shfl_xor(p, 4, 16);
        p += __shfl_xor(p, 2, 16);
        p += __shfl_xor(p, 1, 16);
        if (n == 0) y[idx] = __builtin_fmaf(Dh, uu, p);      // + D*u
    }
}

// ---------------------------------------------------------------------------
extern "C" void kernel_launch(void* const* d_in, const int* in_sizes, int n_in,
                              void* d_out, int out_size, void* d_ws, size_t ws_size,
                              hipStream_t stream) {
    const float* u     = (const float*)d_in[0];  // (B, L, H)
    const float* A_log = (const float*)d_in[1];  // (H, NS)
    const float* B_p   = (const float*)d_in[2];  // (H, NS)
    const float* C_p   = (const float*)d_in[3];  // (H, NS)
    const float* D_p   = (const float*)d_in[4];  // (H,)
    const float* W_dt  = (const float*)d_in[5];  // (H, H)
    const float* b_dt  = (const float*)d_in[6];  // (H,)
    float* y  = (float*)d_out;                   // (B, L, H)
    float* dt = (float*)d_ws;                    // (B*L, H) scratch, 16 MB

    (void)in_sizes; (void)n_in; (void)out_size; (void)ws_size;

    // GEMM: (MTOT/16)*(H/32) waves, 8 waves (256 threads) per block
    {
        const int totalWaves = (MTOT / 16) * (H / 32);   // 256 * 32 = 8192
        const int wavesPerBlk = 256 / 32;
        const int blocks = (totalWaves + wavesPerBlk - 1) / wavesPerBlk;
        s4_dtproj_wmma<<<blocks, 256, 0, stream>>>(u, W_dt, b_dt, dt);
    }

    // Scan: one lane per (b,h,n) -> 2*1024*16 = 32768 threads
    {
        const int total = BZ * H * NS;
        s4_scan<<<(total + 255) / 256, 256, 0, stream>>>(u, A_log, B_p, C_p, D_p, dt, y);
    }
}